// TAOBAOHEGCN_35132832481408
// MI455X (gfx1250) — compile-verified
//
#include <hip/hip_runtime.h>
#include <hip/hip_bf16.h>
#include <float.h>

typedef __attribute__((ext_vector_type(2))) float v2f;
typedef __attribute__((ext_vector_type(8))) float v8f;

constexpr int CH   = 128;      // channels
constexpr int NND  = 80000;    // nodes per type
constexpr int NEDG = 1000000;  // edges per relation
constexpr int NLBL = 400000;   // link labels

__device__ __forceinline__ float warp_sum(float v) {
#pragma unroll
  for (int i = 16; i >= 1; i >>= 1) v += __shfl_xor(v, i, 32);
  return v;
}
__device__ __forceinline__ float sigmoidf(float x) { return 1.0f / (1.0f + expf(-x)); }

// ---------------- utility ----------------
__global__ void zero_f32(float* __restrict__ p, long n) {
  long i = (long)blockIdx.x * blockDim.x + threadIdx.x;
  long s = (long)gridDim.x * blockDim.x;
  for (; i < n; i += s) p[i] = 0.0f;
}

// ---------------- TopK pooling: score = x @ p / ||p|| ----------------
__global__ __launch_bounds__(256) void score_kernel(const float* __restrict__ x,
                                                    const float* __restrict__ p,
                                                    float* __restrict__ score, int n) {
  int wave = blockIdx.x * (blockDim.x >> 5) + (threadIdx.x >> 5);
  int lane = threadIdx.x & 31;
  if (wave >= n) return;
  float4 p4 = ((const float4*)p)[lane];
  float psq = warp_sum(p4.x * p4.x + p4.y * p4.y + p4.z * p4.z + p4.w * p4.w);
  float4 x4 = ((const float4*)(x + (long)wave * CH))[lane];
  float d = warp_sum(x4.x * p4.x + x4.y * p4.y + x4.z * p4.z + x4.w * p4.w);
  if (lane == 0) score[wave] = d / (sqrtf(psq) + 1e-16f);
}

// exact top-k, descending, smallest-index tie-break (matches jax.lax.top_k)
__global__ __launch_bounds__(1024) void topk_kernel(float* __restrict__ score,
                                                    float* __restrict__ topvals,
                                                    int* __restrict__ topidx, int n, int k) {
  __shared__ float sv[1024];
  __shared__ int si[1024];
  int t = threadIdx.x;
  for (int kk = 0; kk < k; ++kk) {
    float best = -FLT_MAX;
    int bi = 0x7fffffff;
    for (int i = t; i < n; i += 1024) {
      float v = score[i];
      if (v > best || (v == best && i < bi)) { best = v; bi = i; }
    }
    sv[t] = best; si[t] = bi;
    __syncthreads();
    for (int off = 512; off >= 1; off >>= 1) {
      if (t < off) {
        float ov = sv[t + off]; int oi = si[t + off];
        if (ov > sv[t] || (ov == sv[t] && oi < si[t])) { sv[t] = ov; si[t] = oi; }
      }
      __syncthreads();
    }
    if (t == 0) {
      topvals[kk] = sv[0];
      topidx[kk] = si[0];
      score[si[0]] = -FLT_MAX;   // remove winner; score is ours to destroy
    }
    __syncthreads();
  }
}

__global__ void xtilde_kernel(const float* __restrict__ x, const float* __restrict__ topvals,
                              const int* __restrict__ topidx, float* __restrict__ Xt) {
  int r = blockIdx.x, c = threadIdx.x;
  Xt[r * CH + c] = x[(long)topidx[r] * CH + c] * tanhf(topvals[r]);
}

// ---------------- one GRU step evolving the GCN weight ----------------
__global__ __launch_bounds__(CH) void gru_kernel(const float* __restrict__ Xt,
                                                 const float* __restrict__ W0,
                                                 const float* __restrict__ Wih,
                                                 const float* __restrict__ Whh,
                                                 const float* __restrict__ bih,
                                                 const float* __restrict__ bhh,
                                                 float* __restrict__ Wout) {
  __shared__ float sXt[CH], sW0[CH];
  int r = blockIdx.x, c = threadIdx.x;
  sXt[c] = Xt[r * CH + c];
  sW0[c] = W0[r * CH + c];
  __syncthreads();
  float ir = bih[c], iz = bih[c + CH], in_ = bih[c + 2 * CH];
  float hr = bhh[c], hz = bhh[c + CH], hn = bhh[c + 2 * CH];
  const float* wir = Wih + (long)c * CH;
  const float* wiz = Wih + (long)(c + CH) * CH;
  const float* win = Wih + (long)(c + 2 * CH) * CH;
  const float* whr = Whh + (long)c * CH;
  const float* whz = Whh + (long)(c + CH) * CH;
  const float* whn = Whh + (long)(c + 2 * CH) * CH;
  for (int k = 0; k < CH; ++k) {
    float a = sXt[k], h = sW0[k];
    ir += a * wir[k]; iz += a * wiz[k]; in_ += a * win[k];
    hr += h * whr[k]; hz += h * whz[k]; hn += h * whn[k];
  }
  float rg = sigmoidf(ir + hr);
  float zg = sigmoidf(iz + hz);
  float ng = tanhf(in_ + rg * hn);
  Wout[r * CH + c] = (1.0f - zg) * ng + zg * sW0[c];
}

// ---------------- xw = x @ W via V_WMMA_F32_16X16X4_F32 ----------------
// One wave owns a 16-row strip and all 8 N-tiles (8 x v8f accumulators).
// Fragment layouts per CDNA5 ISA 7.12.2 (fp32, wave32).
__global__ __launch_bounds__(256) void gemm_wmma(const float* __restrict__ A,
                                                 const float* __restrict__ B,
                                                 float* __restrict__ D, int M) {
  int w = threadIdx.x >> 5;
  int lane = threadIdx.x & 31;
  int half = lane >> 4;
  int l = lane & 15;
  int m0 = (blockIdx.x * 8 + w) * 16;
  if (m0 >= M) return;                        // wave-uniform; EXEC stays all-ones
  v8f acc[8] = {};
  const float* arow = A + (long)(m0 + l) * CH + 2 * half;  // A 16x4: lanes0-15 K0/K1, lanes16-31 K2/K3
  for (int k = 0; k < CH; k += 4) {
    v2f a;
    a.x = arow[k];
    a.y = arow[k + 1];
#pragma unroll
    for (int nt = 0; nt < 8; ++nt) {
      const float* bp = B + (long)(k + 2 * half) * CH + nt * 16 + l;  // B 4x16 row-per-VGPR, half-split
      v2f b;
      b.x = bp[0];
      b.y = bp[CH];
      acc[nt] = __builtin_amdgcn_wmma_f32_16x16x4_f32(
          false, a, false, b, (short)0, acc[nt], false, false);
    }
  }
#pragma unroll
  for (int nt = 0; nt < 8; ++nt)
#pragma unroll
    for (int v = 0; v < 8; ++v)
      D[(long)(m0 + v + 8 * half) * CH + nt * 16 + l] = acc[nt][v];  // C/D: VGPR v -> M=v / v+8
}

// ---------------- GCN edge processing ----------------
__global__ void degree_kernel(const int* __restrict__ edge, float* __restrict__ deg_s,
                              float* __restrict__ deg_d, int e) {
  int i = blockIdx.x * blockDim.x + threadIdx.x;
  if (i >= e) return;
  atomicAdd(&deg_s[edge[i]], 1.0f);
  atomicAdd(&deg_d[edge[e + i]], 1.0f);
}

__global__ __launch_bounds__(256) void scatter_kernel(const float* __restrict__ xw,
                                                      const int* __restrict__ edge,
                                                      const float* __restrict__ deg_s,
                                                      const float* __restrict__ deg_d,
                                                      float* __restrict__ h, int e) {
  int wave = blockIdx.x * 8 + (threadIdx.x >> 5);
  int lane = threadIdx.x & 31;
  if (wave >= e) return;
  int r = edge[wave], c = edge[e + wave];
  float ds = deg_s[r], dd = deg_d[c];
  float is = ds > 0.0f ? rsqrtf(fmaxf(ds, 1.0f)) : 0.0f;
  float id = dd > 0.0f ? rsqrtf(fmaxf(dd, 1.0f)) : 0.0f;
  float nrm = is * id;
  float4 v = ((const float4*)(xw + (long)r * CH))[lane];
  float* out = h + (long)c * CH + lane * 4;
  atomicAdd(out + 0, v.x * nrm);
  atomicAdd(out + 1, v.y * nrm);
  atomicAdd(out + 2, v.z * nrm);
  atomicAdd(out + 3, v.w * nrm);
}

// ---------------- link scorer: relu fused, W_post collapsed to one vector ----------------
__global__ __launch_bounds__(256) void score_links(const float* __restrict__ hu,
                                                   const float* __restrict__ hi,
                                                   const int* __restrict__ lbl,
                                                   const float* __restrict__ Wp,
                                                   const float* __restrict__ bp,
                                                   float* __restrict__ out, int L) {
  int wave = blockIdx.x * 8 + (threadIdx.x >> 5);
  int lane = threadIdx.x & 31;
  if (wave >= L) return;
  int s = lbl[wave], d = lbl[L + wave];
  float4 a = ((const float4*)(hu + (long)s * CH))[lane];
  float4 b = ((const float4*)(hi + (long)d * CH))[lane];
  int c0 = lane * 4;
  float acc =
      fmaxf(a.x, 0.f) * fmaxf(b.x, 0.f) * (Wp[2 * c0 + 0] + Wp[2 * c0 + 1]) +
      fmaxf(a.y, 0.f) * fmaxf(b.y, 0.f) * (Wp[2 * (c0 + 1)] + Wp[2 * (c0 + 1) + 1]) +
      fmaxf(a.z, 0.f) * fmaxf(b.z, 0.f) * (Wp[2 * (c0 + 2)] + Wp[2 * (c0 + 2) + 1]) +
      fmaxf(a.w, 0.f) * fmaxf(b.w, 0.f) * (Wp[2 * (c0 + 3)] + Wp[2 * (c0 + 3) + 1]);
  acc = warp_sum(acc);
  if (lane == 0) out[wave] = acc + bp[0] + bp[1];
}

extern "C" void kernel_launch(void* const* d_in, const int* in_sizes, int n_in,
                              void* d_out, int out_size, void* d_ws, size_t ws_size,
                              hipStream_t stream) {
  const float* xs[2]  = { (const float*)d_in[0], (const float*)d_in[1] };   // x_user, x_item
  const int*   eg[2]  = { (const int*)d_in[2], (const int*)d_in[3] };       // edge_ui, edge_iu
  const int*   elbl   = (const int*)d_in[4];
  const float* p[2]   = { (const float*)d_in[5],  (const float*)d_in[11] };
  const float* W0[2]  = { (const float*)d_in[6],  (const float*)d_in[12] };
  const float* Wih[2] = { (const float*)d_in[7],  (const float*)d_in[13] };
  const float* Whh[2] = { (const float*)d_in[8],  (const float*)d_in[14] };
  const float* bih[2] = { (const float*)d_in[9],  (const float*)d_in[15] };
  const float* bhh[2] = { (const float*)d_in[10], (const float*)d_in[16] };
  const float* Wp = (const float*)d_in[17];
  const float* bp = (const float*)d_in[18];

  char* ws = (char*)d_ws;
  size_t off = 0;
  auto alloc = [&](size_t bytes) -> void* {
    void* pt = ws + off;
    off += (bytes + 255) & ~(size_t)255;
    return pt;
  };
  float* h_item  = (float*)alloc((size_t)NND * CH * 4);
  float* h_user  = (float*)alloc((size_t)NND * CH * 4);
  float* xw      = (float*)alloc((size_t)NND * CH * 4);
  float* scorebf = (float*)alloc((size_t)NND * 4);
  float* deg_s   = (float*)alloc((size_t)NND * 4);
  float* deg_d   = (float*)alloc((size_t)NND * 4);
  float* Xt      = (float*)alloc((size_t)CH * CH * 4);
  float* Wev     = (float*)alloc((size_t)CH * CH * 4);
  float* topvals = (float*)alloc((size_t)CH * 4);
  int*   topidx  = (int*)alloc((size_t)CH * 4);
  float* hout[2] = { h_item, h_user };  // rel 0 (ui) -> item dst, rel 1 (iu) -> user dst

  for (int rel = 0; rel < 2; ++rel) {
    zero_f32<<<2048, 256, 0, stream>>>(hout[rel], (long)NND * CH);
    zero_f32<<<256, 256, 0, stream>>>(deg_s, NND);
    zero_f32<<<256, 256, 0, stream>>>(deg_d, NND);
    score_kernel<<<NND / 8, 256, 0, stream>>>(xs[rel], p[rel], scorebf, NND);
    topk_kernel<<<1, 1024, 0, stream>>>(scorebf, topvals, topidx, NND, CH);
    xtilde_kernel<<<CH, CH, 0, stream>>>(xs[rel], topvals, topidx, Xt);
    gru_kernel<<<CH, CH, 0, stream>>>(Xt, W0[rel], Wih[rel], Whh[rel], bih[rel], bhh[rel], Wev);
    gemm_wmma<<<NND / 128, 256, 0, stream>>>(xs[rel], Wev, xw, NND);
    degree_kernel<<<(NEDG + 255) / 256, 256, 0, stream>>>(eg[rel], deg_s, deg_d, NEDG);
    scatter_kernel<<<(NEDG + 7) / 8, 256, 0, stream>>>(xw, eg[rel], deg_s, deg_d, hout[rel], NEDG);
  }
  score_links<<<(NLBL + 7) / 8, 256, 0, stream>>>(h_user, h_item, elbl, Wp, bp,
                                                  (float*)d_out, NLBL);
}